// LTRelationshipHead_48430051230299
// MI455X (gfx1250) — compile-verified
//
#include <hip/hip_runtime.h>
#include <math.h>

#define DIMX  256
#define OUTD  256
#define NQL   512
#define NQT   512
#define NPTS  20
#define NCLS  10
#define PI_F  3.14159265358979323846f

typedef __attribute__((ext_vector_type(2))) float v2f;
typedef __attribute__((ext_vector_type(8))) float v8f;

// ---------------------------------------------------------------------------
// Prep: lane_in[q][0:256] = mean_p lane_feats[q][p][:]; lane_in[q][256:512] = angle(q)
// ---------------------------------------------------------------------------
__global__ void prep_lane_kernel(const float* __restrict__ lane_feats,
                                 const float* __restrict__ lane_coords,
                                 float* __restrict__ lane_in) {
    const int q = blockIdx.x;
    const int d = threadIdx.x;

    const float* base = lane_feats + (size_t)q * NPTS * DIMX + d;
    float s = 0.f;
#pragma unroll
    for (int p = 0; p < NPTS; ++p) s += base[p * DIMX];
    s *= (1.0f / (float)NPTS);

    const float vx = lane_coords[(q * NPTS + 16) * 2 + 0] - lane_coords[(q * NPTS + 3) * 2 + 0];
    const float vy = lane_coords[(q * NPTS + 16) * 2 + 1] - lane_coords[(q * NPTS + 3) * 2 + 1];
    const float nrm = sqrtf(vx * vx + vy * vy) + 1e-8f;
    const float c = fminf(fmaxf(vx / nrm, -1.0f), 1.0f);
    const float ang = PI_F - acosf(c);

    lane_in[(size_t)q * (2 * DIMX) + d] = s;
    lane_in[(size_t)q * (2 * DIMX) + DIMX + d] = ang;
}

// ---------------------------------------------------------------------------
// Prep: te_in[q][0:256] = te_feats[q]; te_in[q][256:512] = 2*sigmoid(max_c cls[q][c])
// ---------------------------------------------------------------------------
__global__ void prep_te_kernel(const float* __restrict__ te_feats,
                               const float* __restrict__ te_cls,
                               float* __restrict__ te_in) {
    const int q = blockIdx.x;
    const int d = threadIdx.x;

    float m = -INFINITY;
#pragma unroll
    for (int c = 0; c < NCLS; ++c) m = fmaxf(m, te_cls[q * NCLS + c]);
    const float score = 2.0f / (1.0f + expf(-m));

    te_in[(size_t)q * (2 * DIMX) + d] = te_feats[(size_t)q * DIMX + d];
    te_in[(size_t)q * (2 * DIMX) + DIMX + d] = score;
}

// ---------------------------------------------------------------------------
// f32 WMMA GEMM with compile-time shapes:  C[M,N] = [relu]( A@B + bias )
// Each wave owns one 16-row M tile x TWO 16-col N tiles (A fragment reused
// for 2 v_wmma issues per K-step). Constant N/K -> immediate-offset loads,
// no per-iteration 64-bit address math.
//
// VGPR layouts (CDNA5 ISA 7.12.2):
//   A (16x4):  lane m holds (K=k0,k0+1); lane m+16 holds (K=k0+2,k0+3)
//   B (4x16):  lane n holds rows (K=k0,k0+1) at col n; upper half K+2,K+3
//   C/D:       vgpr r -> row r (+8 for lanes 16-31); col = lane%16
// Grid sized exactly; guard is wave-uniform => EXEC all-ones at every WMMA.
// ---------------------------------------------------------------------------
template <int M, int N, int K, int RELU, int HASBIAS>
__global__ void gemm_wmma_f32_kernel(const float* __restrict__ A,
                                     const float* __restrict__ B,
                                     const float* __restrict__ bias,
                                     float* __restrict__ C) {
    constexpr int NTP = N / 32;              // N tile-pairs per M row
    const int wave = (int)((blockIdx.x * blockDim.x + threadIdx.x) >> 5);
    const int lane = (int)(threadIdx.x & 31);
    const int tm   = (wave / NTP) << 4;
    const int tn   = (wave % NTP) << 5;      // 32 columns per wave
    if (tm >= M) return;                     // wave-uniform

    const int half = lane >> 4;              // 0: K+{0,1}, 1: K+{2,3}
    const int sub  = lane & 15;

    const float* Arow  = A + (size_t)(tm + sub) * K + 2 * half;   // + k
    const float* Bbase = B + (size_t)(2 * half) * N + tn + sub;   // + k*N [+16]

    v8f acc0 = {};
    v8f acc1 = {};
#pragma unroll 8
    for (int k = 0; k < K; k += 4) {
        v2f a = *(const v2f*)(Arow + k);     // 8B-aligned (K%4==0, 2*half+k even)
        v2f b0, b1;
        b0.x = Bbase[k * N];                 // B[k+2h  ][tn+sub]
        b0.y = Bbase[k * N + N];             // B[k+2h+1][tn+sub]
        b1.x = Bbase[k * N + 16];            // B[k+2h  ][tn+16+sub]
        b1.y = Bbase[k * N + N + 16];        // B[k+2h+1][tn+16+sub]
        acc0 = __builtin_amdgcn_wmma_f32_16x16x4_f32(false, a, false, b0,
                                                     (short)0, acc0, false, false);
        acc1 = __builtin_amdgcn_wmma_f32_16x16x4_f32(false, a, false, b1,
                                                     (short)0, acc1, false, false);
    }

    float bv0 = 0.0f, bv1 = 0.0f;
    if (HASBIAS) { bv0 = bias[tn + sub]; bv1 = bias[tn + 16 + sub]; }

    float* Cp = C + (size_t)(tm + 8 * half) * N + tn + sub;
#pragma unroll
    for (int r = 0; r < 8; ++r) {
        float v0 = acc0[r] + bv0;
        float v1 = acc1[r] + bv1;
        if (RELU) { v0 = fmaxf(v0, 0.0f); v1 = fmaxf(v1, 0.0f); }
        Cp[r * N]      = v0;
        Cp[r * N + 16] = v1;
    }
}

// ---------------------------------------------------------------------------
// Pairwise head: out[i,j] = b2 + sum_k relu(la[i,k] + tb[j,k] + b1[k]) * w2[k]
// b1 is folded into the staged la tile. Row stride 257 words -> conflict-free
// LDS (stride 256 would alias all 16 tb rows onto one bank).
// ---------------------------------------------------------------------------
__global__ void pairwise_kernel(const float* __restrict__ la,
                                const float* __restrict__ tb,
                                const float* __restrict__ b1,
                                const float* __restrict__ w2,
                                const float* __restrict__ b2,
                                float* __restrict__ out) {
    __shared__ float sla[16][OUTD + 1];
    __shared__ float stb[16][OUTD + 1];
    __shared__ float sw2[OUTD];

    const int i0 = blockIdx.y << 4;
    const int j0 = blockIdx.x << 4;
    const int tid = threadIdx.x;

    for (int idx = tid; idx < 16 * OUTD; idx += 256) {
        const int r = idx >> 8;
        const int k = idx & (OUTD - 1);
        sla[r][k] = la[(size_t)(i0 + r) * OUTD + k] + b1[k];   // fold bias here
        stb[r][k] = tb[(size_t)(j0 + r) * OUTD + k];
    }
    sw2[tid] = w2[tid];
    __syncthreads();

    const int ti = tid >> 4;
    const int tj = tid & 15;
    float acc = 0.0f;
#pragma unroll 8
    for (int k = 0; k < OUTD; ++k) {
        const float s = sla[ti][k] + stb[tj][k];
        acc = fmaf(fmaxf(s, 0.0f), sw2[k], acc);
    }
    out[(size_t)(i0 + ti) * NQT + (j0 + tj)] = acc + b2[0];
}

// ---------------------------------------------------------------------------
extern "C" void kernel_launch(void* const* d_in, const int* in_sizes, int n_in,
                              void* d_out, int out_size, void* d_ws, size_t ws_size,
                              hipStream_t stream) {
    const float* lane_feats  = (const float*)d_in[0];
    const float* lane_coords = (const float*)d_in[1];
    const float* te_feats    = (const float*)d_in[2];
    const float* te_cls      = (const float*)d_in[3];
    const float* o1_w1 = (const float*)d_in[4];
    const float* o1_b1 = (const float*)d_in[5];
    const float* o1_w2 = (const float*)d_in[6];
    const float* o1_b2 = (const float*)d_in[7];
    const float* o2_w1 = (const float*)d_in[8];
    const float* o2_b1 = (const float*)d_in[9];
    const float* o2_w2 = (const float*)d_in[10];
    const float* o2_b2 = (const float*)d_in[11];
    const float* cls_w1 = (const float*)d_in[12];
    const float* cls_b1 = (const float*)d_in[13];
    const float* cls_w2 = (const float*)d_in[14];
    const float* cls_b2 = (const float*)d_in[15];

    // workspace (floats): inbuf(512*512) | h(512*256) | emb(512*256) | la | tb  = 3 MB
    float* ws    = (float*)d_ws;
    float* inbuf = ws;
    float* h     = inbuf + (size_t)NQL * 2 * DIMX;
    float* emb   = h   + (size_t)NQL * DIMX;
    float* la    = emb + (size_t)NQL * OUTD;
    float* tb    = la  + (size_t)NQL * OUTD;

    const float* w1a = cls_w1;                 // rows 0..255
    const float* w1b = cls_w1 + OUTD * OUTD;   // rows 256..511

    // waves = (M/16)*(N/32) = 32*8 = 256; 8 waves per 256-thread block -> 32 blocks
    const dim3 gblk(32), gthr(256);

    // --- lane branch ---
    prep_lane_kernel<<<NQL, DIMX, 0, stream>>>(lane_feats, lane_coords, inbuf);
    gemm_wmma_f32_kernel<NQL, DIMX, 2 * DIMX, 1, 1>
        <<<gblk, gthr, 0, stream>>>(inbuf, o1_w1, o1_b1, h);
    gemm_wmma_f32_kernel<NQL, OUTD, DIMX, 0, 1>
        <<<gblk, gthr, 0, stream>>>(h, o1_w2, o1_b2, emb);
    gemm_wmma_f32_kernel<NQL, OUTD, OUTD, 0, 0>
        <<<gblk, gthr, 0, stream>>>(emb, w1a, nullptr, la);

    // --- te branch (reuses inbuf/h/emb, stream-ordered) ---
    prep_te_kernel<<<NQT, DIMX, 0, stream>>>(te_feats, te_cls, inbuf);
    gemm_wmma_f32_kernel<NQT, DIMX, 2 * DIMX, 1, 1>
        <<<gblk, gthr, 0, stream>>>(inbuf, o2_w1, o2_b1, h);
    gemm_wmma_f32_kernel<NQT, OUTD, DIMX, 0, 1>
        <<<gblk, gthr, 0, stream>>>(h, o2_w2, o2_b2, emb);
    gemm_wmma_f32_kernel<NQT, OUTD, OUTD, 0, 0>
        <<<gblk, gthr, 0, stream>>>(emb, w1b, nullptr, tb);

    // --- pairwise head ---
    pairwise_kernel<<<dim3(NQT / 16, NQL / 16), 256, 0, stream>>>(
        la, tb, cls_b1, cls_w2, cls_b2, (float*)d_out);
}